// Mamba2D_61684320305339
// MI455X (gfx1250) — compile-verified
//
#include <hip/hip_runtime.h>
#include <math.h>

typedef __attribute__((ext_vector_type(2))) float v2f;
typedef __attribute__((ext_vector_type(8))) float v8f;
typedef __attribute__((ext_vector_type(4))) unsigned int u32x4;
typedef __attribute__((ext_vector_type(8))) int i32x8;
typedef __attribute__((ext_vector_type(4))) int i32x4;

#define B_   4
#define C_   128
#define L_   4096
#define BL_  (B_ * L_)
#define DIN  256
#define DST  16
#define NDBC 288   // 256 delta + 16 B + 16 C

// ---------------------------------------------------------------------------
// fp32 WMMA: D(16x16) = A(16x4) * B(4x16) + C  -> v_wmma_f32_16x16x4_f32
// Layouts (ISA 7.12.2, wave32):
//   A: lane holds row m = lane&15, K pair {2*(lane>>4), 2*(lane>>4)+1}
//   B: lane holds col n = lane&15, same K pairing per VGPR
//   C/D: VGPR i holds M = i + 8*(lane>>4), N = lane&15
// ---------------------------------------------------------------------------
__device__ __forceinline__ v8f wmma4(v2f a, v2f b, v8f c) {
  return __builtin_amdgcn_wmma_f32_16x16x4_f32(false, a, false, b, (short)0, c,
                                               false, false);
}

__device__ __forceinline__ v8f vzero8() {
  v8f z;
#pragma unroll
  for (int i = 0; i < 8; ++i) z[i] = 0.0f;
  return z;
}

// Make the TDM's LDS writes visible to the optimizer: without this the
// compiler sees no store to the shared array and folds its loads to undef.
__device__ __forceinline__ void lds_written_barrier(const float* p) {
  asm volatile("" : : "v"(p) : "memory");
}

// ---------------------------------------------------------------------------
// TDM: DMA a [rows x k_elems] fp32 row-major panel from global into LDS at
// the shared-array address lds. D# bitfields per CDNA5 ISA 8.3/8.4 (2D
// tensor, groups 2/3 zeroed). 6-arg builtin (clang-23 lane). One wave issues;
// completion via s_wait_tensorcnt, then workgroup barrier releases consumers.
// ---------------------------------------------------------------------------
__device__ __forceinline__ void tdm_load_panel(const float* gsrc, const float* lds,
                                               unsigned k_elems, unsigned rows) {
  unsigned long long ga = (unsigned long long)(const void*)gsrc;
  // Generic pointer to LDS: low 32 bits are the LDS byte offset (aperture map).
  unsigned lds_off = (unsigned)(unsigned long long)(const void*)lds;
  u32x4 g0;
  g0[0] = 1u;                                            // count=1 (valid user D#)
  g0[1] = lds_off;                                       // lds_addr (bytes)
  g0[2] = (unsigned)(ga & 0xFFFFFFFFu);                  // global_addr[31:0]
  g0[3] = (unsigned)((ga >> 32) & 0x01FFFFFFu) | (2u << 30);  // addr[56:32] | type=2
  i32x8 g1;
  g1[0] = (int)(2u << 16);                               // data_size=4B
  g1[1] = (int)((k_elems & 0xFFFFu) << 16);              // tensor_dim0 lo16
  g1[2] = (int)(((k_elems >> 16) & 0xFFFFu) |            // tensor_dim0 hi16
                ((rows & 0xFFFFu) << 16));               // tensor_dim1 lo16
  g1[3] = (int)(((rows >> 16) & 0xFFFFu) |               // tensor_dim1 hi16
                ((k_elems & 0xFFFFu) << 16));            // tile_dim0
  g1[4] = (int)(rows & 0xFFFFu);                         // tile_dim1 (tile_dim2=0)
  g1[5] = (int)k_elems;                                  // tensor_dim0_stride lo32
  g1[6] = 0;                                             // stride hi / dim1_stride
  g1[7] = 0;
  i32x4 z4; z4[0] = z4[1] = z4[2] = z4[3] = 0;
  i32x8 z8;
#pragma unroll
  for (int i = 0; i < 8; ++i) z8[i] = 0;
  __builtin_amdgcn_tensor_load_to_lds(g0, g1, z4, z4, z8, 0);
}

// ---------------------------------------------------------------------------
// K1: W2[o][c] = sum_m in_proj_w[o][m] * proj_w[m][c]   (512 x 128)
// ---------------------------------------------------------------------------
__global__ __launch_bounds__(256) void build_W2(const float* __restrict__ ipw,
                                                const float* __restrict__ pw,
                                                float* __restrict__ W2) {
  int idx = blockIdx.x * 256 + threadIdx.x;  // 512*128
  int c = idx & 127, o = idx >> 7;
  float s = 0.0f;
  for (int m = 0; m < 128; ++m) s += ipw[o * 128 + m] * pw[m * 128 + c];
  W2[idx] = s;
}

// ---------------------------------------------------------------------------
// K2: Wdbc (288 x 256): rows 0..255 = dt_proj_w(256x8) @ x_proj_w[0:8,:]
//                       rows 256..287 = x_proj_w[8:40,:]  (B then C rows)
// ---------------------------------------------------------------------------
__global__ __launch_bounds__(256) void build_Wdbc(const float* __restrict__ xpw,
                                                  const float* __restrict__ dpw,
                                                  float* __restrict__ Wdbc) {
  int idx = blockIdx.x * 256 + threadIdx.x;  // 288*256
  int c = idx & 255, r = idx >> 8;
  float v;
  if (r < 256) {
    v = 0.0f;
#pragma unroll
    for (int j = 0; j < 8; ++j) v += dpw[r * 8 + j] * xpw[j * 256 + c];
  } else {
    v = xpw[(8 + (r - 256)) * 256 + c];
  }
  Wdbc[r * 256 + c] = v;
}

// ---------------------------------------------------------------------------
// K3: xz[b][l][j] = sum_c x[b][c][l] * W2[j][c]
// Block = 8 M-tiles x one 64-wide N-group; B-panel (64x128) staged via TDM.
// ---------------------------------------------------------------------------
__global__ __launch_bounds__(256) void gemm_xz(const float* __restrict__ X,
                                               const float* __restrict__ W2,
                                               float* __restrict__ xz) {
  __shared__ float Bs[64 * 128];  // 32 KB
  int b    = blockIdx.x >> 8;
  int ngrp = (blockIdx.x >> 5) & 7;
  int mgrp = blockIdx.x & 31;
  int waveid = threadIdx.x >> 5;
  int lane   = threadIdx.x & 31;
  int row = lane & 15, half = lane >> 4;
  int m0 = (mgrp * 8 + waveid) * 16;
  int n0 = ngrp * 64;

  if (waveid == 0) {
    tdm_load_panel(W2 + (size_t)n0 * C_, Bs, C_, 64);
    __builtin_amdgcn_s_wait_tensorcnt(0);
  }
  __syncthreads();
  lds_written_barrier(Bs);

  const float* Xb = X + (size_t)b * C_ * L_;
  v8f acc[4];
#pragma unroll
  for (int t = 0; t < 4; ++t) acc[t] = vzero8();

  for (int k0 = 0; k0 < C_; k0 += 4) {
    int c0 = k0 + 2 * half;
    v2f a;
    a.x = Xb[(size_t)c0 * L_ + m0 + row];  // x is (c-major, l contiguous)
    a.y = Xb[(size_t)(c0 + 1) * L_ + m0 + row];
#pragma unroll
    for (int t = 0; t < 4; ++t) {
      const float2 bv = *(const float2*)(Bs + (t * 16 + row) * C_ + c0);
      v2f bb; bb.x = bv.x; bb.y = bv.y;
      acc[t] = wmma4(a, bb, acc[t]);
    }
  }
#pragma unroll
  for (int t = 0; t < 4; ++t) {
    int col = n0 + t * 16 + row;
#pragma unroll
    for (int i = 0; i < 8; ++i) {
      int m = i + 8 * half;
      xz[((size_t)b * L_ + m0 + m) * 512 + col] = acc[t][i];
    }
  }
}

// ---------------------------------------------------------------------------
// K4: depthwise causal conv (k=4) + bias + SiLU:  xc[b][l][d]
// ---------------------------------------------------------------------------
__global__ __launch_bounds__(256) void conv_silu(const float* __restrict__ xz,
                                                 const float* __restrict__ cw,
                                                 const float* __restrict__ cb,
                                                 float* __restrict__ xc) {
  int idx = blockIdx.x * 256 + threadIdx.x;  // BL_ * DIN
  int d  = idx & (DIN - 1);
  int bl = idx >> 8;
  int b = bl >> 12, l = bl & (L_ - 1);
  float acc = cb[d];
#pragma unroll
  for (int k = 0; k < 4; ++k) {
    int ls = l - 3 + k;
    if (ls >= 0) acc += cw[d * 4 + k] * xz[((size_t)b * L_ + ls) * 512 + d];
  }
  xc[(size_t)bl * DIN + d] = acc / (1.0f + expf(-acc));
}

// ---------------------------------------------------------------------------
// K5: dbc = xc(BL x 256) @ Wdbc^T(256 x 288); cols<256: softplus(.+dt_proj_b)
// Block = 8 M-tiles x one 48-wide N-group; B-panel (48x256) staged via TDM.
// ---------------------------------------------------------------------------
__global__ __launch_bounds__(256) void gemm_dbc(const float* __restrict__ xc,
                                                const float* __restrict__ Wdbc,
                                                const float* __restrict__ dtb,
                                                float* __restrict__ dbc) {
  __shared__ float Bs[48 * 256];  // 48 KB
  int ngrp = blockIdx.x % 6;
  int mgrp = blockIdx.x / 6;      // 0..127
  int waveid = threadIdx.x >> 5;
  int lane   = threadIdx.x & 31;
  int row = lane & 15, half = lane >> 4;
  int m0 = (mgrp * 8 + waveid) * 16;
  int n0 = ngrp * 48;

  if (waveid == 0) {
    tdm_load_panel(Wdbc + (size_t)n0 * DIN, Bs, DIN, 48);
    __builtin_amdgcn_s_wait_tensorcnt(0);
  }
  __syncthreads();
  lds_written_barrier(Bs);

  v8f acc[3];
#pragma unroll
  for (int t = 0; t < 3; ++t) acc[t] = vzero8();

  for (int k0 = 0; k0 < DIN; k0 += 4) {
    int kk = k0 + 2 * half;
    const float2 av = *(const float2*)(xc + (size_t)(m0 + row) * DIN + kk);
    v2f a; a.x = av.x; a.y = av.y;
#pragma unroll
    for (int t = 0; t < 3; ++t) {
      const float2 bv = *(const float2*)(Bs + (t * 16 + row) * DIN + kk);
      v2f bb; bb.x = bv.x; bb.y = bv.y;
      acc[t] = wmma4(a, bb, acc[t]);
    }
  }
#pragma unroll
  for (int t = 0; t < 3; ++t) {
    int col = n0 + t * 16 + row;
#pragma unroll
    for (int i = 0; i < 8; ++i) {
      int m = i + 8 * half;
      float v = acc[t][i];
      if (col < DIN) {  // delta path: softplus(v + bias)
        v += dtb[col];
        v = (v > 20.0f) ? v : log1pf(expf(v));
      }
      dbc[(size_t)(m0 + m) * NDBC + col] = v;
    }
  }
}

// ---------------------------------------------------------------------------
// K6: selective scan. 32 waves; one lane = one channel d, 16 states in regs.
// B/C (32 contiguous floats at dbc[...,256:288]) broadcast via __shfl.
// ---------------------------------------------------------------------------
__global__ __launch_bounds__(32) void scan_kernel(const float* __restrict__ dbc,
                                                  const float* __restrict__ xc,
                                                  const float* __restrict__ xz,
                                                  const float* __restrict__ A_log,
                                                  const float* __restrict__ Dp,
                                                  float* __restrict__ yg) {
  int wid  = blockIdx.x;        // 0..31
  int b    = wid >> 3;
  int dg   = wid & 7;
  int lane = threadIdx.x;       // 0..31
  int d    = dg * 32 + lane;

  float Av[DST], h[DST];
#pragma unroll
  for (int n = 0; n < DST; ++n) {
    Av[n] = -expf(A_log[d * DST + n]);
    h[n]  = 0.0f;
  }
  float Dd = Dp[d];

  for (int l = 0; l < L_; ++l) {
    size_t base = (size_t)b * L_ + l;
    float delta = dbc[base * NDBC + d];
    float xv    = xc[base * DIN + d];
    float g     = xz[base * 512 + DIN + d];
    float bc    = dbc[base * NDBC + DIN + lane];  // lanes 0..15: B, 16..31: C
    float dx = delta * xv;
    float y  = 0.0f;
#pragma unroll
    for (int n = 0; n < DST; ++n) {
      float Bn = __shfl(bc, n, 32);
      float Cn = __shfl(bc, n + 16, 32);
      h[n] = expf(delta * Av[n]) * h[n] + dx * Bn;
      y += h[n] * Cn;
    }
    float yv = y + xv * Dd;
    yg[base * DIN + d] = yv * (g / (1.0f + expf(-g)));
  }
}

// ---------------------------------------------------------------------------
// K7: out[b][c][l] = x[b][c][l] + sum_k yg[b][l][k] * out_proj_w[c][k]
// Block = 8 M-tiles x one 64-wide N-group; B-panel (64x256) staged via TDM.
// D-tile rows are contiguous in l -> float4 residual-add epilogue.
// ---------------------------------------------------------------------------
__global__ __launch_bounds__(256) void gemm_out(const float* __restrict__ Y,
                                                const float* __restrict__ Wout,
                                                const float* __restrict__ X,
                                                float* __restrict__ out) {
  __shared__ float Bs[64 * 256];  // 64 KB
  int ngrp = blockIdx.x & 1;
  int mgrp = blockIdx.x >> 1;     // 0..127
  int waveid = threadIdx.x >> 5;
  int lane   = threadIdx.x & 31;
  int row = lane & 15, half = lane >> 4;
  int m0 = (mgrp * 8 + waveid) * 16;
  int n0 = ngrp * 64;
  int b = m0 >> 12, l0 = m0 & (L_ - 1);

  if (waveid == 0) {
    tdm_load_panel(Wout + (size_t)n0 * DIN, Bs, DIN, 64);
    __builtin_amdgcn_s_wait_tensorcnt(0);
  }
  __syncthreads();
  lds_written_barrier(Bs);

  v8f acc[4];
#pragma unroll
  for (int t = 0; t < 4; ++t) acc[t] = vzero8();

  for (int k0 = 0; k0 < DIN; k0 += 4) {
    int kk = k0 + 2 * half;
    const float2 av = *(const float2*)(Y + (size_t)(m0 + row) * DIN + kk);
    v2f a; a.x = av.x; a.y = av.y;
#pragma unroll
    for (int t = 0; t < 4; ++t) {
      const float2 bv = *(const float2*)(Bs + (t * 16 + row) * DIN + kk);
      v2f bb; bb.x = bv.x; bb.y = bv.y;
      acc[t] = wmma4(a, bb, acc[t]);
    }
  }
#pragma unroll
  for (int t = 0; t < 4; ++t) {
    int c = n0 + t * 16 + row;
    size_t base = ((size_t)b * C_ + c) * L_ + l0 + 8 * half;
    float4 r0 = *(const float4*)(X + base);
    float4 r1 = *(const float4*)(X + base + 4);
    float4 o0, o1;
    o0.x = acc[t][0] + r0.x; o0.y = acc[t][1] + r0.y;
    o0.z = acc[t][2] + r0.z; o0.w = acc[t][3] + r0.w;
    o1.x = acc[t][4] + r1.x; o1.y = acc[t][5] + r1.y;
    o1.z = acc[t][6] + r1.z; o1.w = acc[t][7] + r1.w;
    *(float4*)(out + base)     = o0;
    *(float4*)(out + base + 4) = o1;
  }
}

// ---------------------------------------------------------------------------
extern "C" void kernel_launch(void* const* d_in, const int* in_sizes, int n_in,
                              void* d_out, int out_size, void* d_ws, size_t ws_size,
                              hipStream_t stream) {
  const float* x         = (const float*)d_in[0];
  const float* proj_w    = (const float*)d_in[1];
  const float* in_proj_w = (const float*)d_in[2];
  const float* conv_w    = (const float*)d_in[3];
  const float* conv_b    = (const float*)d_in[4];
  const float* x_proj_w  = (const float*)d_in[5];
  const float* dt_proj_w = (const float*)d_in[6];
  const float* dt_proj_b = (const float*)d_in[7];
  const float* A_log     = (const float*)d_in[8];
  const float* Dp        = (const float*)d_in[9];
  const float* out_proj_w= (const float*)d_in[10];
  float* out = (float*)d_out;

  float* ws   = (float*)d_ws;
  float* W2   = ws;                               // 512*128
  float* Wdbc = W2 + 512 * 128;                   // 288*256
  float* xz   = Wdbc + NDBC * DIN;                // BL*512
  float* xc   = xz + (size_t)BL_ * 512;           // BL*256
  float* dbc  = xc + (size_t)BL_ * DIN;           // BL*288
  float* yg   = dbc + (size_t)BL_ * NDBC;         // BL*256

  build_W2  <<<256, 256, 0, stream>>>(in_proj_w, proj_w, W2);
  build_Wdbc<<<288, 256, 0, stream>>>(x_proj_w, dt_proj_w, Wdbc);
  gemm_xz   <<<1024, 256, 0, stream>>>(x, W2, xz);
  conv_silu <<<(BL_ * DIN) / 256, 256, 0, stream>>>(xz, conv_w, conv_b, xc);
  gemm_dbc  <<<768, 256, 0, stream>>>(xc, Wdbc, dt_proj_b, dbc);
  scan_kernel<<<32, 32, 0, stream>>>(dbc, xc, xz, A_log, Dp, yg);
  gemm_out  <<<256, 256, 0, stream>>>(yg, out_proj_w, x, out);
}